// VectorQuantizer_55705725829901
// MI455X (gfx1250) — compile-verified
//
#include <hip/hip_runtime.h>

typedef float v2f __attribute__((ext_vector_type(2)));
typedef float v8f __attribute__((ext_vector_type(8)));

#define KC   1024            // codebook entries
#define DC   64              // embedding dim (== C)
#define SPB  16384           // L*H*W = 16*32*32
#define NVEC 65536           // B*L*H*W rows
#define TOT  4194304         // B*C*L*H*W elements
#define NWAVES (NVEC / 32)   // 2048 waves, 32 rows each
#define WS_PART 1024         // ws[0..1023]=||cb_k||^2, ws[1024..3071]=loss partials

// ---- kernel 1: codebook squared norms ---------------------------------
__global__ __launch_bounds__(256) void vq_norms(const float* __restrict__ cb,
                                                float* __restrict__ ws) {
  int k = blockIdx.x * 256 + threadIdx.x;
  if (k < KC) {
    const float* r = cb + (size_t)k * DC;
    float s = 0.f;
#pragma unroll
    for (int c = 0; c < DC; ++c) { float v = r[c]; s += v * v; }
    ws[k] = s;
  }
}

// ---- kernel 2: distances via f32 WMMA + argmin + gather + loss partials
__global__ __launch_bounds__(128) void vq_main(const float* __restrict__ x,
                                               const float* __restrict__ cb,
                                               float* __restrict__ out,
                                               float* __restrict__ ws) {
  __shared__ int s_idx[4 * 32];

  const int lane = threadIdx.x & 31;
  const int wv   = threadIdx.x >> 5;
  const int half = lane >> 4;      // 0: K+0/K+1 slice, 1: K+2/K+3 slice
  const int m    = lane & 15;      // row within 16-row subtile / code column
  const int wgid = blockIdx.x * 4 + wv;
  const int row0 = wgid * 32;      // this wave owns rows [row0, row0+32)

  // lane's two x-rows (n -> inputs[b, :, s] with channel stride SPB)
  const int n0 = row0 + m;
  const int n1 = row0 + 16 + m;
  const float* xb0 = x + (size_t)(n0 >> 14) * ((size_t)DC * SPB) + (n0 & 16383);
  const float* xb1 = x + (size_t)(n1 >> 14) * ((size_t)DC * SPB) + (n1 & 16383);

  // A tiles (x rows, pre-scaled by -2) in WMMA 16x4-f32 layout:
  // lane m, half h, step j:  a[2j] = x[m][4j+2h],  a[2j+1] = x[m][4j+2h+1]
  float a0[32], a1[32];
#pragma unroll
  for (int j = 0; j < 16; ++j) {
    const int c0 = 4 * j + 2 * half;
    a0[2*j+0] = -2.0f * xb0[(size_t)(c0)     * SPB];
    a0[2*j+1] = -2.0f * xb0[(size_t)(c0 + 1) * SPB];
    a1[2*j+0] = -2.0f * xb1[(size_t)(c0)     * SPB];
    a1[2*j+1] = -2.0f * xb1[(size_t)(c0 + 1) * SPB];
  }

  float mv0[8], mv1[8];
  int   mi0[8], mi1[8];
#pragma unroll
  for (int v = 0; v < 8; ++v) { mv0[v] = 3e38f; mv1[v] = 3e38f; mi0[v] = 0; mi1[v] = 0; }

  for (int t = 0; t < KC / 16; ++t) {
    const int code = t * 16 + m;                 // this lane's code column
    const float cn = ws[code];                   // ||cb_code||^2
    const float* cbp = cb + (size_t)code * DC + 2 * half;
    v8f acc0 = {cn, cn, cn, cn, cn, cn, cn, cn}; // D = ||cb||^2 - 2 x.cb
    v8f acc1 = acc0;
#pragma unroll
    for (int j = 0; j < 16; ++j) {
      v2f b = *(const v2f*)(cbp + 4 * j);        // B: cb[code][4j+2h], [4j+2h+1]
      v2f av0; av0.x = a0[2*j]; av0.y = a0[2*j+1];
      v2f av1; av1.x = a1[2*j]; av1.y = a1[2*j+1];
      acc0 = __builtin_amdgcn_wmma_f32_16x16x4_f32(false, av0, false, b,
                                                   (short)0, acc0, false, false);
      acc1 = __builtin_amdgcn_wmma_f32_16x16x4_f32(false, av1, false, b,
                                                   (short)0, acc1, false, false);
    }
#pragma unroll
    for (int v = 0; v < 8; ++v) {
      float d0 = acc0[v]; if (d0 < mv0[v]) { mv0[v] = d0; mi0[v] = code; }
      float d1 = acc1[v]; if (d1 < mv1[v]) { mv1[v] = d1; mi1[v] = code; }
    }
  }

  // cross-lane argmin over the 16 code columns (within each 16-lane half),
  // tie-break on smaller index to match jnp.argmin first-occurrence
#pragma unroll
  for (int mask = 1; mask <= 8; mask <<= 1) {
#pragma unroll
    for (int v = 0; v < 8; ++v) {
      float ov = __shfl_xor(mv0[v], mask, 32);
      int   oi = __shfl_xor(mi0[v], mask, 32);
      if (ov < mv0[v] || (ov == mv0[v] && oi < mi0[v])) { mv0[v] = ov; mi0[v] = oi; }
      ov = __shfl_xor(mv1[v], mask, 32);
      oi = __shfl_xor(mi1[v], mask, 32);
      if (ov < mv1[v] || (ov == mv1[v] && oi < mi1[v])) { mv1[v] = ov; mi1[v] = oi; }
    }
  }

  // stage per-row winners in LDS (slot v holds row half*8+v of each subtile)
  if (m == 0) {
#pragma unroll
    for (int v = 0; v < 8; ++v) {
      s_idx[wv * 32 + half * 8 + v]      = mi0[v];
      s_idx[wv * 32 + 16 + half * 8 + v] = mi1[v];
    }
  }
  __syncthreads();

  // encoding indices (as float), coalesced 32-lane store
  float* out_idx = out + (TOT + 1);
  out_idx[row0 + lane] = (float)s_idx[wv * 32 + lane];

  // gather winning codebook rows, write quantized output, accumulate loss
  const int my0 = s_idx[wv * 32 + m];
  const int my1 = s_idx[wv * 32 + 16 + m];
  const float* q0 = cb + (size_t)my0 * DC;
  const float* q1 = cb + (size_t)my1 * DC;
  float* ob0 = out + (size_t)(n0 >> 14) * ((size_t)DC * SPB) + (n0 & 16383);
  float* ob1 = out + (size_t)(n1 >> 14) * ((size_t)DC * SPB) + (n1 & 16383);
  float lsum = 0.f;
#pragma unroll
  for (int cc = 0; cc < 32; ++cc) {
    const int c = 2 * cc + half;
    const size_t o = (size_t)c * SPB;
    float qa = q0[c], xa = xb0[o];
    ob0[o] = qa;
    float d = qa - xa; lsum += d * d;
    float qb = q1[c], xbv = xb1[o];
    ob1[o] = qb;
    d = qb - xbv; lsum += d * d;
  }
#pragma unroll
  for (int mask = 1; mask < 32; mask <<= 1) lsum += __shfl_xor(lsum, mask, 32);
  if (lane == 0) ws[WS_PART + wgid] = lsum;     // deterministic: no atomics
}

// ---- kernel 3: fixed-order loss reduction -----------------------------
__global__ __launch_bounds__(256) void vq_loss(const float* __restrict__ ws,
                                               float* __restrict__ out) {
  __shared__ float red[256];
  float s = 0.f;
  for (int i = threadIdx.x; i < NWAVES; i += 256) s += ws[WS_PART + i];
  red[threadIdx.x] = s;
  __syncthreads();
  for (int st = 128; st > 0; st >>= 1) {
    if ((int)threadIdx.x < st) red[threadIdx.x] += red[threadIdx.x + st];
    __syncthreads();
  }
  if (threadIdx.x == 0)
    out[TOT] = 1.25f * red[0] * (1.0f / (float)TOT);   // (1+0.25)*MSE
}

extern "C" void kernel_launch(void* const* d_in, const int* in_sizes, int n_in,
                              void* d_out, int out_size, void* d_ws, size_t ws_size,
                              hipStream_t stream) {
  const float* x  = (const float*)d_in[0];   // inputs [4,64,16,32,32] fp32
  const float* cb = (const float*)d_in[1];   // codebook [1024,64] fp32
  float* out = (float*)d_out;                // [quantized | loss | indices]
  float* ws  = (float*)d_ws;

  vq_norms<<<dim3(4), dim3(256), 0, stream>>>(cb, ws);
  vq_main<<<dim3(NVEC / 32 / 4), dim3(128), 0, stream>>>(x, cb, out, ws);
  vq_loss<<<dim3(1), dim3(256), 0, stream>>>(ws, out);
}